// CRD_80272938762449
// MI455X (gfx1250) — compile-verified
//
#include <hip/hip_runtime.h>
#include <math.h>

// ---------------- problem constants (from reference) ----------------
#define S_DIM   2048
#define T_DIM   2048
#define FEAT    128
#define BATCH   64
#define K1      16385          // NCE_N + 1
#define N_DATA  100000
#define INV_T   (1.0f / 0.07f)
#define M_CONST (16384.0f / 100000.0f)
#define EPS_C   1e-7f

typedef float v2f __attribute__((ext_vector_type(2)));
typedef float v8f __attribute__((ext_vector_type(8)));

// ---------------- workspace layout (float offsets) ------------------
#define WS_HRAW   0                         // [2][64][128]
#define WS_HNORM  (WS_HRAW + 2*BATCH*FEAT)  // [2][64][128]
#define WS_OUT    (WS_HNORM + 2*BATCH*FEAT) // [2][64][16385]
#define WS_PART   (WS_OUT + 2*BATCH*K1)     // [2][64][8] = 1024
#define WS_Z      (WS_PART + 1024)          // [2]
#define WS_BPART  (WS_Z + 2)                // [512]

// =====================================================================
// Kernel 1: h_raw = feat @ W^T + bias, via V_WMMA_F32_16X16X4_F32.
// 64 waves total: wave -> (side, 16x16 output tile). M=64, N=128, K=2048.
// Full blocks, no divergence: EXEC is all-ones around WMMA as required.
// =====================================================================
__global__ void kEmbed(const float* __restrict__ featS, const float* __restrict__ featT,
                       const float* __restrict__ wS, const float* __restrict__ bS,
                       const float* __restrict__ wT, const float* __restrict__ bT,
                       float* __restrict__ hraw) {
    const int wave = (blockIdx.x * blockDim.x + threadIdx.x) >> 5;  // 0..63
    const int lane = threadIdx.x & 31;
    const int side = wave >> 5;          // 32 tiles per side (4 M-tiles x 8 N-tiles)
    const int tile = wave & 31;
    const int m0 = (tile >> 3) * 16;
    const int n0 = (tile & 7) * 16;
    const int g = lane >> 4;             // lane half
    const int r = lane & 15;

    const float* A    = side ? featT : featS;   // [64, 2048] row-major
    const float* W    = side ? wT    : wS;      // [128, 2048] row-major (B = W^T)
    const float* bias = side ? bT    : bS;
    float* out = hraw + side * (BATCH * FEAT);

    const float* arow = A + (size_t)(m0 + r) * S_DIM;   // A[m=r + m0][*]
    const float* wrow = W + (size_t)(n0 + r) * S_DIM;   // B[k][n=r+n0] = W[n][k]

    v8f c = {};
    #pragma unroll 4
    for (int k = 0; k < S_DIM; k += 4) {
        // ISA A-layout 16x4: VGPR0 = K=(2g), VGPR1 = K=(2g+1), rows striped on lanes
        v2f a, b;
        a[0] = arow[k + 2 * g];
        a[1] = arow[k + 2 * g + 1];
        // B-layout 4x16 (transposed analogue): rows 2g, 2g+1 striped across lane halves
        b[0] = wrow[k + 2 * g];
        b[1] = wrow[k + 2 * g + 1];
        c = __builtin_amdgcn_wmma_f32_16x16x4_f32(
                /*neg_a=*/false, a, /*neg_b=*/false, b,
                /*c_mod=*/(short)0, c, /*reuse_a=*/false, /*reuse_b=*/false);
    }
    // D-layout: VGPR j -> row (m0 + j + 8*g), col (n0 + r)
    const float bn = bias[n0 + r];
    #pragma unroll
    for (int j = 0; j < 8; ++j) {
        out[(size_t)(m0 + j + 8 * g) * FEAT + n0 + r] = c[j] + bn;
    }
}

// =====================================================================
// Kernel 2: row-wise L2 normalize. One wave per row (128 floats = 4/lane),
// wave32 butterfly reduction.
// =====================================================================
__global__ void kNorm(const float* __restrict__ hraw, float* __restrict__ hnorm) {
    const int wave = (blockIdx.x * blockDim.x + threadIdx.x) >> 5;  // 0..127
    const int lane = threadIdx.x & 31;
    const float4 v = ((const float4*)(hraw + (size_t)wave * FEAT))[lane];
    float s = v.x * v.x + v.y * v.y + v.z * v.z + v.w * v.w;
    #pragma unroll
    for (int off = 16; off; off >>= 1) s += __shfl_xor(s, off, 32);
    const float inv = 1.0f / sqrtf(s);
    float4 o;
    o.x = v.x * inv; o.y = v.y * inv; o.z = v.z * inv; o.w = v.w * inv;
    ((float4*)(hnorm + (size_t)wave * FEAT))[lane] = o;
}

// =====================================================================
// Kernel 3 (dominant, L2-BW bound): gathered-row dot + exp.
// Block = (side, b); 8 waves stride over k. One 512B row per wave per k:
// global_load_b128 per lane, dot4, 5-step butterfly, expf, store.
// Per-wave partial sums of exp() written for the deterministic z reduction.
// =====================================================================
__global__ void kGather(const float* __restrict__ hnorm,
                        const int* __restrict__ sampleIdx,
                        const float* __restrict__ memV1, const float* __restrict__ memV2,
                        float* __restrict__ outv, float* __restrict__ partial) {
    const int side = blockIdx.x >> 6;           // 0: hs vs mem_v2, 1: ht vs mem_v1
    const int b    = blockIdx.x & 63;
    const int wave = threadIdx.x >> 5;
    const int lane = threadIdx.x & 31;

    const float* h = hnorm + ((size_t)side * BATCH + b) * FEAT;
    const float* mem = side ? memV1 : memV2;
    const float4 h4 = ((const float4*)h)[lane];
    const int* sIdx = sampleIdx + (size_t)b * K1;
    float* orow = outv + ((size_t)side * BATCH + b) * K1;

    float lsum = 0.0f;
    for (int k = wave; k < K1; k += 8) {
        const int idx = sIdx[k];
        const float4 v = ((const float4*)(mem + (size_t)idx * FEAT))[lane];
        float p = v.x * h4.x + v.y * h4.y + v.z * h4.z + v.w * h4.w;
        #pragma unroll
        for (int off = 16; off; off >>= 1) p += __shfl_xor(p, off, 32);
        const float e = expf(p * INV_T);
        if (lane == 0) orow[k] = e;
        lsum += e;                      // identical across lanes post-butterfly
    }
    if (lane == 0) partial[((size_t)side * BATCH + b) * 8 + wave] = lsum;
}

// =====================================================================
// Kernel 4: deterministic tree-reduce 512 partials per side -> z.
// =====================================================================
__global__ void kZ(const float* __restrict__ partial, float* __restrict__ z) {
    __shared__ float s0[256], s1[256];
    const int t = threadIdx.x;
    s0[t] = partial[t]       + partial[t + 256];
    s1[t] = partial[512 + t] + partial[768 + t];
    __syncthreads();
    for (int off = 128; off; off >>= 1) {
        if (t < off) { s0[t] += s0[t + off]; s1[t] += s1[t + off]; }
        __syncthreads();
    }
    if (t == 0) {
        const float scale = (float)N_DATA / ((float)BATCH * (float)K1); // mean * N
        z[0] = s0[0] * scale;
        z[1] = s1[0] * scale;
    }
}

// =====================================================================
// Kernel 5: loss terms, grid-stride with per-block tree reduce (fixed order).
// =====================================================================
__global__ void kLoss(const float* __restrict__ outv, const float* __restrict__ z,
                      float* __restrict__ blockpart) {
    const int total = 2 * BATCH * K1;
    float acc = 0.0f;
    for (int i = blockIdx.x * blockDim.x + threadIdx.x; i < total;
         i += gridDim.x * blockDim.x) {
        const int side = i / (BATCH * K1);
        const int rem  = i - side * (BATCH * K1);
        const int k    = rem % K1;
        const float x  = outv[i] / z[side];
        const float denom = x + M_CONST + EPS_C;
        acc += (k == 0) ? logf(x / denom) : logf(M_CONST / denom);
    }
    __shared__ float s[256];
    s[threadIdx.x] = acc;
    __syncthreads();
    for (int off = 128; off; off >>= 1) {
        if (threadIdx.x < off) s[threadIdx.x] += s[threadIdx.x + off];
        __syncthreads();
    }
    if (threadIdx.x == 0) blockpart[blockIdx.x] = s[0];
}

// =====================================================================
// Kernel 6: final reduce of 512 block partials -> scalar loss.
// =====================================================================
__global__ void kFinalize(const float* __restrict__ blockpart, float* __restrict__ out) {
    __shared__ float s[256];
    const int t = threadIdx.x;
    s[t] = blockpart[t] + blockpart[t + 256];
    __syncthreads();
    for (int off = 128; off; off >>= 1) {
        if (t < off) s[t] += s[t + off];
        __syncthreads();
    }
    if (t == 0) out[0] = -s[0] / (float)BATCH;
}

// =====================================================================
extern "C" void kernel_launch(void* const* d_in, const int* in_sizes, int n_in,
                              void* d_out, int out_size, void* d_ws, size_t ws_size,
                              hipStream_t stream) {
    (void)in_sizes; (void)n_in; (void)out_size; (void)ws_size;
    // setup_inputs order:
    const float* feat_s     = (const float*)d_in[0];
    const float* feat_t     = (const float*)d_in[1];
    // d_in[2] = idx (unused: sample_idx already has positives in column 0)
    const int*   sample_idx = (const int*)d_in[3];
    const float* w_s        = (const float*)d_in[4];
    const float* b_s        = (const float*)d_in[5];
    const float* w_t        = (const float*)d_in[6];
    const float* b_t        = (const float*)d_in[7];
    const float* mem_v1     = (const float*)d_in[8];
    const float* mem_v2     = (const float*)d_in[9];

    float* ws    = (float*)d_ws;
    float* hraw  = ws + WS_HRAW;
    float* hnorm = ws + WS_HNORM;
    float* outv  = ws + WS_OUT;
    float* part  = ws + WS_PART;
    float* z     = ws + WS_Z;
    float* bpart = ws + WS_BPART;
    float* loss  = (float*)d_out;

    kEmbed   <<<  8, 256, 0, stream>>>(feat_s, feat_t, w_s, b_s, w_t, b_t, hraw);
    kNorm    <<< 16, 256, 0, stream>>>(hraw, hnorm);
    kGather  <<<128, 256, 0, stream>>>(hnorm, sample_idx, mem_v1, mem_v2, outv, part);
    kZ       <<<  1, 256, 0, stream>>>(part, z);
    kLoss    <<<512, 256, 0, stream>>>(outv, z, bpart);
    kFinalize<<<  1, 256, 0, stream>>>(bpart, loss);
}